// GshellHelper_9801115370056
// MI455X (gfx1250) — compile-verified
//
// Marching tetrahedra (GshellHelper) for MI455X / gfx1250, compile-only loop.
// Memory-bound pipeline: edge-key build -> 5-pass 8-bit LSD radix sort (key =
// (lo<<20)|hi, order-equivalent to lo*NV+hi; stable ranking via wave32 ballot
// multi-split) -> unique/inverse (sorted, matches jnp.unique semantics) ->
// crossing flags + prefix-sum compaction -> vert interpolation scatter ->
// per-tet triangle-table faces.
// CDNA5-specific: TDM tensor_load_to_lds + s_wait_tensorcnt (faces kernel),
// async global->LDS b128 staging + s_wait_asynccnt (scan tiles), wave32 ballot
// ranking, LDS scans, global_prefetch in the sort histogram pass.
// Roofline: ~1.5 GB total HBM traffic @ 23.3 TB/s => ~65 us floor; all global
// access patterns are lane-contiguous (>=128B per wave request).

#include <hip/hip_runtime.h>
#include <cstdint>
#include <cstddef>

#define THREADS 256            // 8 wave32 waves per block
#define SCAN_IPT 16
#define SCAN_TILE (THREADS * SCAN_IPT)   // 4096 dwords
#define SCAN_ROW 20            // padded LDS row stride (16B-aligned, few conflicts)
#define SORT_IPT 8
#define SORT_TILE (THREADS * SORT_IPT)   // 2048
#define RADIX_BITS 8
#define RADIX 256
#define RADIX_PASSES 5         // 40 bits >= 39 significant key bits
#define KEY_SHIFT 20           // key = (lo << 20) | hi  (valid while NV < 2^20)
#define NWAVES (THREADS / 32)

typedef unsigned int u32x4 __attribute__((ext_vector_type(4)));
typedef int          i32x4 __attribute__((ext_vector_type(4)));
typedef int          i32x8 __attribute__((ext_vector_type(8)));

#if defined(__has_builtin)
#if __has_builtin(__builtin_amdgcn_tensor_load_to_lds) && \
    __has_builtin(__builtin_amdgcn_s_wait_tensorcnt)
#define HAVE_TDM 1
#endif
#if __has_builtin(__builtin_amdgcn_ballot_w32)
#define HAVE_BALLOT_W32 1
#endif
#if __has_builtin(__builtin_amdgcn_global_load_async_to_lds_b128) && \
    __has_builtin(__builtin_amdgcn_s_wait_asynccnt)
#define HAVE_ASYNC_LDS 1
#endif
#endif

__device__ __forceinline__ unsigned ballot32(bool p) {
#ifdef HAVE_BALLOT_W32
  return __builtin_amdgcn_ballot_w32(p);
#else
  return (unsigned)__ballot(p);
#endif
}

#ifdef HAVE_ASYNC_LDS
// Builtin signature (from hipcc diagnostic): param0 is v4i32 in AS1 (global,
// printed as "__device__"); LDS side is the AS3 counterpart.
typedef __attribute__((address_space(1))) i32x4 as1_i32x4;
typedef __attribute__((address_space(3))) i32x4 as3_i32x4;
// 16B async copy global->LDS, tracked on ASYNCcnt (no VGPR round-trip).
__device__ __forceinline__ void async_copy_b128(const void* g, void* l) {
  as1_i32x4* gp = (as1_i32x4*)(uintptr_t)g;
  as3_i32x4* lp = (as3_i32x4*)(uint32_t)(uintptr_t)l;  // low 32 bits = LDS offset
  __builtin_amdgcn_global_load_async_to_lds_b128(gp, lp, 0, 0);
}
#endif

__constant__ int c_edge_pairs[12] = {0,1, 0,2, 0,3, 1,2, 1,3, 2,3};
__constant__ int c_num_tri[16]    = {0,1,1,2,1,2,2,1,1,2,2,1,2,1,1,0};
__constant__ int c_tri_table[16][6] = {
  {-1,-1,-1,-1,-1,-1},{1,0,2,-1,-1,-1},{4,0,3,-1,-1,-1},{1,4,2,1,3,4},
  {3,1,5,-1,-1,-1},{2,3,0,2,5,3},{1,4,0,1,5,4},{4,2,5,-1,-1,-1},
  {4,5,2,-1,-1,-1},{4,1,0,4,5,1},{3,2,0,3,5,2},{1,3,5,-1,-1,-1},
  {4,1,2,4,3,1},{3,0,4,-1,-1,-1},{2,0,1,-1,-1,-1},{-1,-1,-1,-1,-1,-1}};

// ---------------------------------------------------------------- TDM helper
#ifdef HAVE_TDM
// 1-row tile load: tileElems 4-byte elements from gptr into LDS at ldsOff.
// tensorElems = remaining in-bounds elements (TDM zero-fills OOB reads).
__device__ inline void tdm_load_tile(uint32_t ldsOff, const void* gptr,
                                     uint32_t tensorElems, uint32_t tileElems) {
  uint64_t ga = (uint64_t)(uintptr_t)gptr;
  u32x4 g0;
  g0.x = 1u;                                            // count = 1 (valid D#)
  g0.y = ldsOff;                                        // lds_addr [63:32]
  g0.z = (uint32_t)ga;                                  // global_addr lo
  g0.w = (uint32_t)((ga >> 32) & 0x01FFFFFFu) | (2u << 30); // addr hi | type=2
  i32x8 g1;
  g1[0] = (int)(2u << 16);                              // data_size = 4 bytes
  g1[1] = (int)((tensorElems & 0xFFFFu) << 16);         // tensor_dim0 [15:0]
  g1[2] = (int)(((tensorElems >> 16) & 0xFFFFu) | (1u << 16)); // td0 hi | td1=1
  g1[3] = (int)((tileElems & 0xFFFFu) << 16);           // tile_dim0
  g1[4] = 1;                                            // tile_dim1 = 1 row
  g1[5] = (int)tensorElems;                             // tensor_dim0_stride lo
  g1[6] = 0;
  g1[7] = 0;
  i32x4 z4; z4[0] = 0; z4[1] = 0; z4[2] = 0; z4[3] = 0;
#if defined(__clang_major__) && (__clang_major__ >= 23)
  i32x8 z8; z8[0]=0; z8[1]=0; z8[2]=0; z8[3]=0; z8[4]=0; z8[5]=0; z8[6]=0; z8[7]=0;
  __builtin_amdgcn_tensor_load_to_lds(g0, g1, z4, z4, z8, 0);
#else
  __builtin_amdgcn_tensor_load_to_lds(g0, g1, z4, z4, 0);
#endif
}
#endif

// ------------------------------------------------------------------- kernels
__global__ __launch_bounds__(THREADS)
void build_codes(const int* __restrict__ tet, uint64_t* __restrict__ keys,
                 uint32_t* __restrict__ pay, int nt) {
  int e = blockIdx.x * THREADS + threadIdx.x;
  int total = nt * 6;
  if (e >= total) return;
  int t = e / 6, j = e - t * 6;
  int a = tet[(size_t)t * 4 + c_edge_pairs[2 * j + 0]];
  int b = tet[(size_t)t * 4 + c_edge_pairs[2 * j + 1]];
  int lo = a < b ? a : b;
  int hi = a < b ? b : a;
  keys[e] = ((uint64_t)(uint32_t)lo << KEY_SHIFT) | (uint32_t)hi;
  pay[e]  = (uint32_t)e;
}

// 256-bin histogram; wave32 ballot multi-split so only one LDS atomic per
// distinct digit per 32-element round. Reads are lane-contiguous (coalesced).
__global__ __launch_bounds__(THREADS)
void radix_hist(const uint64_t* __restrict__ keys, uint32_t* __restrict__ hist,
                int n, int shift, int numBlocks) {
  __shared__ uint32_t h[RADIX];
  int tid = threadIdx.x, blk = blockIdx.x;
  int wave = tid >> 5, lane = tid & 31;
  h[tid] = 0u;
  __syncthreads();
  long waveBase = (long)blk * SORT_TILE + (long)wave * (32 * SORT_IPT);
  __builtin_prefetch(keys + waveBase + SORT_TILE, 0, 0);   // global_prefetch_b8
#pragma unroll
  for (int k = 0; k < SORT_IPT; k++) {
    long i = waveBase + k * 32 + lane;
    bool valid = i < n;
    uint64_t kk = valid ? keys[i] : 0;
    int d = (int)((kk >> shift) & (RADIX - 1));
    unsigned mask = ballot32(valid);
#pragma unroll
    for (int bit = 0; bit < RADIX_BITS; bit++) {
      unsigned bb = ballot32(((d >> bit) & 1) != 0);
      mask &= ((d >> bit) & 1) ? bb : ~bb;
    }
    if (valid && lane == (int)__builtin_ctz(mask))
      atomicAdd(&h[d], (uint32_t)__builtin_popcount(mask));
  }
  __syncthreads();
  hist[(size_t)tid * numBlocks + blk] = h[tid];
}

// In-place exclusive scan by a single 256-thread block (chunked Hillis-Steele).
// Only used for small arrays (block-sum vectors).
__global__ __launch_bounds__(THREADS)
void scan_exclusive_block(uint32_t* a, int n) {
  __shared__ uint32_t s[THREADS];
  __shared__ uint32_t carry;
  int tid = threadIdx.x;
  if (tid == 0) carry = 0u;
  __syncthreads();
  for (int base = 0; base < n; base += THREADS) {
    int i = base + tid;
    uint32_t v = (i < n) ? a[i] : 0u;
    s[tid] = v;
    __syncthreads();
    for (int off = 1; off < THREADS; off <<= 1) {
      uint32_t t = (tid >= off) ? s[tid - off] : 0u;
      __syncthreads();
      s[tid] += t;
      __syncthreads();
    }
    uint32_t excl = s[tid] - v + carry;
    if (i < n) a[i] = excl;
    uint32_t total = s[THREADS - 1];
    __syncthreads();
    if (tid == 0) carry += total;
    __syncthreads();
  }
}

// Stable 8-bit scatter. Elements are wave-major: idx = blkBase + wave*256 +
// round*32 + lane, and positions are (wave-exclusive, prior rounds, in-round
// lane rank) => stable. In-round ranking via wave32 ballot multi-split.
__global__ __launch_bounds__(THREADS)
void radix_scatter(const uint64_t* __restrict__ keysIn, const uint32_t* __restrict__ payIn,
                   uint64_t* __restrict__ keysOut, uint32_t* __restrict__ payOut,
                   const uint32_t* __restrict__ histEx, int n, int shift, int numBlocks) {
  __shared__ uint32_t whist[RADIX * NWAVES];   // [bin][wave] counts
  __shared__ uint32_t gbase[RADIX];
  int tid = threadIdx.x, blk = blockIdx.x;
  int wave = tid >> 5, lane = tid & 31;
  for (int i = tid; i < RADIX * NWAVES; i += THREADS) whist[i] = 0u;
  gbase[tid] = histEx[(size_t)tid * numBlocks + blk];
  __syncthreads();

  long waveBase = (long)blk * SORT_TILE + (long)wave * (32 * SORT_IPT);
  uint64_t kv[SORT_IPT]; uint32_t pv[SORT_IPT]; int dg[SORT_IPT];
  uint32_t lr[SORT_IPT]; bool vld[SORT_IPT];
  unsigned ltmask = (1u << lane) - 1u;
#pragma unroll
  for (int k = 0; k < SORT_IPT; k++) {
    long i = waveBase + k * 32 + lane;
    bool valid = i < n;
    vld[k] = valid;
    uint64_t kk = valid ? keysIn[i] : 0;
    uint32_t pp = valid ? payIn[i] : 0;
    kv[k] = kk; pv[k] = pp;
    int d = (int)((kk >> shift) & (RADIX - 1));
    dg[k] = d;
    unsigned mask = ballot32(valid);
#pragma unroll
    for (int bit = 0; bit < RADIX_BITS; bit++) {
      unsigned bb = ballot32(((d >> bit) & 1) != 0);
      mask &= ((d >> bit) & 1) ? bb : ~bb;
    }
    uint32_t rank = (uint32_t)__builtin_popcount(mask & ltmask);
    uint32_t pre = 0u;
    if (valid) pre = whist[d * NWAVES + wave];   // per-wave column: no race
    lr[k] = pre + rank;
    if (valid && lane == (int)__builtin_ctz(mask))
      whist[d * NWAVES + wave] = pre + (uint32_t)__builtin_popcount(mask);
  }
  __syncthreads();
  {                                   // exclusive scan across waves, bin = tid
    uint32_t run = 0u;
#pragma unroll
    for (int w = 0; w < NWAVES; w++) {
      uint32_t t = whist[tid * NWAVES + w];
      whist[tid * NWAVES + w] = run;
      run += t;
    }
  }
  __syncthreads();
#pragma unroll
  for (int k = 0; k < SORT_IPT; k++) {
    if (vld[k]) {
      uint32_t pos = gbase[dg[k]] + whist[dg[k] * NWAVES + wave] + lr[k];
      keysOut[pos] = kv[k];
      payOut[pos]  = pv[k];
    }
  }
}

__global__ __launch_bounds__(THREADS)
void head_flags(const uint64_t* __restrict__ S, uint32_t* __restrict__ flags, int n) {
  int i = blockIdx.x * THREADS + threadIdx.x;
  if (i >= n) return;
  flags[i] = (i == 0 || S[i] != S[i - 1]) ? 1u : 0u;
}

// Tile-level scan (stage 1 of two-level large scan). exclusive!=0 -> exclusive.
// Tiles are staged through LDS so all global traffic is lane-contiguous b128
// chunks (async global->LDS on CDNA5 when available); the per-thread serial
// scan then runs out of a padded LDS layout (row stride SCAN_ROW dwords keeps
// rows 16B-aligned with only 2-way bank conflicts). Safe in-place.
__global__ __launch_bounds__(THREADS)
void scan_tile(const uint32_t* __restrict__ in, uint32_t* __restrict__ out,
               uint32_t* __restrict__ blockSums, int n, int exclusive) {
  __shared__ uint32_t stage[THREADS * SCAN_ROW];  // 20KB padded tile
  __shared__ uint32_t s[THREADS];
  int tid = threadIdx.x, blk = blockIdx.x;
  long tileBase = (long)blk * SCAN_TILE;
  bool full = (tileBase + SCAN_TILE) <= (long)n;

  if (full) {
#ifdef HAVE_ASYNC_LDS
#pragma unroll
    for (int r = 0; r < 4; r++) {
      int c = r * THREADS + tid;            // 16B chunk id within tile
      int t = c >> 2, k0 = (c & 3) * 4;     // target row/col (same row per chunk)
      async_copy_b128(in + tileBase + (long)c * 4, &stage[t * SCAN_ROW + k0]);
    }
    __builtin_amdgcn_s_wait_asynccnt(0);
#else
#pragma unroll
    for (int r = 0; r < 4; r++) {
      int c = r * THREADS + tid;
      int t = c >> 2, k0 = (c & 3) * 4;
      uint4 v4 = *(const uint4*)(in + tileBase + (long)c * 4);
      stage[t * SCAN_ROW + k0 + 0] = v4.x;
      stage[t * SCAN_ROW + k0 + 1] = v4.y;
      stage[t * SCAN_ROW + k0 + 2] = v4.z;
      stage[t * SCAN_ROW + k0 + 3] = v4.w;
    }
#endif
  } else {
    for (int r = 0; r < SCAN_IPT; r++) {
      int jl = r * THREADS + tid;           // lane-contiguous tail load
      long i = tileBase + jl;
      int t = jl >> 4, k = jl & 15;
      stage[t * SCAN_ROW + k] = (i < n) ? in[i] : 0u;
    }
  }
  __syncthreads();

  uint32_t v[SCAN_IPT]; uint32_t run = 0u;
#pragma unroll
  for (int k = 0; k < SCAN_IPT; k++) {
    uint32_t x = stage[tid * SCAN_ROW + k];
    uint32_t pre = run;
    run += x;
    v[k] = exclusive ? pre : run;
  }
  s[tid] = run;
  __syncthreads();
  for (int off = 1; off < THREADS; off <<= 1) {
    uint32_t t = (tid >= off) ? s[tid - off] : 0u;
    __syncthreads();
    s[tid] += t;
    __syncthreads();
  }
  uint32_t excl = s[tid] - run;
#pragma unroll
  for (int k = 0; k < SCAN_IPT; k++) stage[tid * SCAN_ROW + k] = v[k] + excl;
  __syncthreads();

  if (full) {
#pragma unroll
    for (int r = 0; r < 4; r++) {
      int c = r * THREADS + tid;
      int t = c >> 2, k0 = (c & 3) * 4;
      uint4 v4;
      v4.x = stage[t * SCAN_ROW + k0 + 0];
      v4.y = stage[t * SCAN_ROW + k0 + 1];
      v4.z = stage[t * SCAN_ROW + k0 + 2];
      v4.w = stage[t * SCAN_ROW + k0 + 3];
      *(uint4*)(out + tileBase + (long)c * 4) = v4;
    }
  } else {
    for (int r = 0; r < SCAN_IPT; r++) {
      int jl = r * THREADS + tid;
      long i = tileBase + jl;
      if (i < n) {
        int t = jl >> 4, k = jl & 15;
        out[i] = stage[t * SCAN_ROW + k];
      }
    }
  }
  if (tid == THREADS - 1) blockSums[blk] = s[tid];
}

__global__ __launch_bounds__(THREADS)
void scan_add_offsets(uint32_t* __restrict__ data, const uint32_t* __restrict__ blockOffs, int n) {
  int tid = threadIdx.x, blk = blockIdx.x;
  uint32_t off = blockOffs[blk];
  if (off == 0u) return;
  long base = (long)blk * SCAN_TILE;
  for (int k = 0; k < SCAN_IPT; k++) {
    long i = base + (long)k * THREADS + tid;
    if (i < n) data[i] += off;
  }
}

__global__ __launch_bounds__(THREADS)
void finalize_unique(const uint64_t* __restrict__ S, const uint32_t* __restrict__ P,
                     const uint32_t* __restrict__ uidIncl, uint32_t* __restrict__ inv,
                     uint64_t* __restrict__ uniq, uint32_t* __restrict__ nUp, int n) {
  int i = blockIdx.x * THREADS + threadIdx.x;
  if (i >= n) return;
  uint32_t u = uidIncl[i] - 1u;
  inv[P[i]] = u;
  bool head = (i == 0) || (S[i] != S[i - 1]);
  if (head) uniq[u] = S[i];
  if (i == n - 1) *nUp = uidIncl[i];
}

__global__ __launch_bounds__(THREADS)
void crossing_kernel(const uint64_t* __restrict__ uniq, const uint32_t* __restrict__ nUp,
                     const float* __restrict__ sdf, uint32_t* __restrict__ cross,
                     float* __restrict__ crossOut, int n) {
  int u = blockIdx.x * THREADS + threadIdx.x;
  if (u >= n) return;
  uint32_t nU = *nUp;
  uint32_t c = 0u;
  if ((uint32_t)u < nU) {
    uint64_t key = uniq[u];
    int a = (int)(key >> KEY_SHIFT);
    int b = (int)(key & ((1u << KEY_SHIFT) - 1u));
    c = ((sdf[a] > 0.0f) != (sdf[b] > 0.0f)) ? 1u : 0u;
  }
  cross[u]    = c;
  crossOut[u] = (float)c;
}

__global__ __launch_bounds__(THREADS)
void fill_zero(float* __restrict__ p, long n) {
  long i = (long)blockIdx.x * THREADS + threadIdx.x;
  if (i < n) p[i] = 0.0f;
}

__global__ __launch_bounds__(THREADS)
void verts_kernel(const uint64_t* __restrict__ uniq, const uint32_t* __restrict__ cross,
                  const uint32_t* __restrict__ msum, const float* __restrict__ pos,
                  const float* __restrict__ sdf, float* __restrict__ verts, int n) {
  int u = blockIdx.x * THREADS + threadIdx.x;
  if (u >= n) return;
  if (!cross[u]) return;
  uint64_t key = uniq[u];
  int a = (int)(key >> KEY_SHIFT);
  int b = (int)(key & ((1u << KEY_SHIFT) - 1u));
  float s0 = sdf[a], s1 = sdf[b];
  float denom = s0 - s1;
  float w0 = -s1 / denom;
  float w1 =  s0 / denom;
  uint32_t m = msum[u] - 1u;
  verts[(size_t)m * 3 + 0] = pos[(size_t)a * 3 + 0] * w0 + pos[(size_t)b * 3 + 0] * w1;
  verts[(size_t)m * 3 + 1] = pos[(size_t)a * 3 + 1] * w0 + pos[(size_t)b * 3 + 1] * w1;
  verts[(size_t)m * 3 + 2] = pos[(size_t)a * 3 + 2] * w0 + pos[(size_t)b * 3 + 2] * w1;
}

__global__ __launch_bounds__(THREADS)
void faces_kernel(const int* __restrict__ tet, const float* __restrict__ sdf,
                  const uint32_t* __restrict__ inv, const uint32_t* __restrict__ cross,
                  const uint32_t* __restrict__ msum, float* __restrict__ facesOut,
                  float* __restrict__ ntriOut, int nt) {
  __shared__ int sTet[THREADS * 4];
  int blk = blockIdx.x, tid = threadIdx.x;
  int tileBase = blk * THREADS;
#ifdef HAVE_TDM
  if (tid < 32) {                    // one wave issues the TDM for the block
    uint32_t ldsOff = (uint32_t)(uintptr_t)(void*)sTet;
    const int* gsrc = tet + (size_t)tileBase * 4;
    long remL = (long)nt * 4 - (long)tileBase * 4;
    uint32_t rem = remL > 0 ? (uint32_t)remL : 0u;
    tdm_load_tile(ldsOff, gsrc, rem, THREADS * 4);
    __builtin_amdgcn_s_wait_tensorcnt(0);
  }
#else
  for (int i = tid; i < THREADS * 4; i += THREADS) {
    long gi = (long)tileBase * 4 + i;
    sTet[i] = (gi < (long)nt * 4) ? tet[gi] : 0;
  }
#endif
  __syncthreads();
  int t = tileBase + tid;
  if (t >= nt) return;
  int a = sTet[tid * 4 + 0], b = sTet[tid * 4 + 1];
  int c = sTet[tid * 4 + 2], d = sTet[tid * 4 + 3];
  int idx = (sdf[a] > 0.0f ? 1 : 0) | (sdf[b] > 0.0f ? 2 : 0) |
            (sdf[c] > 0.0f ? 4 : 0) | (sdf[d] > 0.0f ? 8 : 0);
  int ntri = c_num_tri[idx];
  ntriOut[t] = (float)ntri;
  for (int k = 0; k < 6; k++) {
    float val = -1.0f;
    if (k < ntri * 3) {
      int e = c_tri_table[idx][k];
      uint32_t ui = inv[(size_t)t * 6 + e];
      val = cross[ui] ? (float)(msum[ui] - 1u) : -1.0f;
    }
    facesOut[(size_t)t * 6 + k] = val;
  }
}

// ---------------------------------------------------------------- host driver
extern "C" void kernel_launch(void* const* d_in, const int* in_sizes, int n_in,
                              void* d_out, int out_size, void* d_ws, size_t ws_size,
                              hipStream_t stream) {
  const float* pos = (const float*)d_in[0];
  const float* sdf = (const float*)d_in[1];
  const int*   tet = (const int*)d_in[2];
  const int nt = in_sizes[2] / 4;
  const int U  = nt * 6;

  float* out      = (float*)d_out;
  float* vertsOut = out;                               // [U,3]
  float* facesOut = out + (size_t)U * 3;               // [nt,2,3]
  float* ntriOut  = facesOut + (size_t)nt * 6;         // [nt]
  float* crossOut = ntriOut + (size_t)nt;              // [U]

  const int nBlkScan = (U + SCAN_TILE - 1) / SCAN_TILE;
  const int nBlkSort = (U + SORT_TILE - 1) / SORT_TILE;
  const int histN    = RADIX * nBlkSort;
  const int nBlkHist = (histN + SCAN_TILE - 1) / SCAN_TILE;

  char* ws = (char*)d_ws;
  size_t cur = 0;
  auto alloc = [&](size_t bytes) -> void* {
    void* p = ws + cur;
    cur += (bytes + 255) & ~(size_t)255;
    return p;
  };
  uint64_t* keysA = (uint64_t*)alloc((size_t)U * 8);
  uint64_t* keysB = (uint64_t*)alloc((size_t)U * 8);
  uint32_t* payA  = (uint32_t*)alloc((size_t)U * 4);
  uint32_t* payB  = (uint32_t*)alloc((size_t)U * 4);
  uint32_t* invA  = (uint32_t*)alloc((size_t)U * 4);
  uint32_t* uid   = (uint32_t*)alloc((size_t)U * 4);
  uint32_t* flags = (uint32_t*)alloc((size_t)U * 4);
  uint32_t* crossF= (uint32_t*)alloc((size_t)U * 4);
  uint32_t* msum  = (uint32_t*)alloc((size_t)U * 4);
  uint32_t* hist  = (uint32_t*)alloc((size_t)histN * 4);
  uint32_t* bsum  = (uint32_t*)alloc((size_t)(nBlkScan > nBlkHist ? nBlkScan : nBlkHist) * 4);
  uint32_t* nUp   = (uint32_t*)alloc(4);

  const int g256 = (U + THREADS - 1) / THREADS;

  build_codes<<<g256, THREADS, 0, stream>>>(tet, keysA, payA, nt);

  uint64_t* ks = keysA; uint64_t* kd = keysB;
  uint32_t* ps = payA;  uint32_t* pd = payB;
  for (int pass = 0; pass < RADIX_PASSES; pass++) {
    int shift = pass * RADIX_BITS;
    radix_hist<<<nBlkSort, THREADS, 0, stream>>>(ks, hist, U, shift, nBlkSort);
    // global exclusive scan of the 256 x nBlkSort histogram (two-level)
    scan_tile<<<nBlkHist, THREADS, 0, stream>>>(hist, hist, bsum, histN, 1);
    scan_exclusive_block<<<1, THREADS, 0, stream>>>(bsum, nBlkHist);
    scan_add_offsets<<<nBlkHist, THREADS, 0, stream>>>(hist, bsum, histN);
    radix_scatter<<<nBlkSort, THREADS, 0, stream>>>(ks, ps, kd, pd, hist, U, shift, nBlkSort);
    uint64_t* tk = ks; ks = kd; kd = tk;
    uint32_t* tp = ps; ps = pd; pd = tp;
  }
  // 5 (odd) passes: sorted keys in ks==keysB, payload in ps==payB; kd==keysA free

  head_flags<<<g256, THREADS, 0, stream>>>(ks, flags, U);
  scan_tile<<<nBlkScan, THREADS, 0, stream>>>(flags, uid, bsum, U, 0);
  scan_exclusive_block<<<1, THREADS, 0, stream>>>(bsum, nBlkScan);
  scan_add_offsets<<<nBlkScan, THREADS, 0, stream>>>(uid, bsum, U);

  uint64_t* uniq = kd;  // reuse free ping-pong buffer for sorted-unique array
  finalize_unique<<<g256, THREADS, 0, stream>>>(ks, ps, uid, invA, uniq, nUp, U);

  crossing_kernel<<<g256, THREADS, 0, stream>>>(uniq, nUp, sdf, crossF, crossOut, U);
  scan_tile<<<nBlkScan, THREADS, 0, stream>>>(crossF, msum, bsum, U, 0);
  scan_exclusive_block<<<1, THREADS, 0, stream>>>(bsum, nBlkScan);
  scan_add_offsets<<<nBlkScan, THREADS, 0, stream>>>(msum, bsum, U);

  long n3 = (long)U * 3;
  fill_zero<<<(int)((n3 + THREADS - 1) / THREADS), THREADS, 0, stream>>>(vertsOut, n3);
  verts_kernel<<<g256, THREADS, 0, stream>>>(uniq, crossF, msum, pos, sdf, vertsOut, U);

  faces_kernel<<<(nt + THREADS - 1) / THREADS, THREADS, 0, stream>>>(
      tet, sdf, invA, crossF, msum, facesOut, ntriOut, nt);

  (void)n_in; (void)out_size; (void)ws_size;
}